// ROIAlign_30494267801811
// MI455X (gfx1250) — compile-verified
//
#include <hip/hip_runtime.h>
#include <hip/hip_bf16.h>

// ---------------------------------------------------------------------------
// ROI-align 1x1, restructured as per-box pixel-weight map + WMMA weighted
// reduction over the ROI footprint only (~38 MB traffic vs 1 GiB full read).
// One wave32 per (box, 16-channel tile); WMMA f32 16x16x4 with the weight
// vector broadcast across N columns.
// ---------------------------------------------------------------------------

typedef __attribute__((ext_vector_type(2))) float v2f;
typedef __attribute__((ext_vector_type(8))) float v8f;

#define BBOX  256
#define CCH   256
#define HH    64
#define WW    64
#define HWSZ  4096
#define CHUNK 64      // pixels staged per iteration
#define ASTR  68      // padded LDS row stride (floats) -> conflict-free A reads

__global__ __launch_bounds__(32)
void roi_align_wmma_kernel(const float* __restrict__ feat,
                           const float* __restrict__ bbox,
                           float* __restrict__ out)
{
    __shared__ float wmap[HWSZ];          // per-box bilinear weight map (16 KB)
    __shared__ float ftile[16 * ASTR];    // 16 channels x CHUNK pixels (padded)
    __shared__ float wchunk[CHUNK];       // weights for current pixel chunk
    __shared__ int   offs[CHUNK];         // y*64+x offsets for current chunk
    __shared__ int   sbnd[4];             // ymin, ymax, xmin, xmax (inclusive)

    const int lane = threadIdx.x;         // 0..31 (wave32)
    const int b    = blockIdx.y;          // box / batch index
    const int c0   = blockIdx.x * 16;     // channel tile base

    // ---------------- Phase A: build per-box weight map in LDS -------------
    for (int i = lane; i < HWSZ / 4; i += 32)
        ((float4*)wmap)[i] = make_float4(0.f, 0.f, 0.f, 0.f);
    if (lane == 0) { sbnd[0] = 64; sbnd[1] = -1; sbnd[2] = 64; sbnd[3] = -1; }
    __syncthreads();

    // Box clamping exactly as in the reference
    const float nb0 = bbox[b * 4 + 0], nb1 = bbox[b * 4 + 1];
    const float nb2 = bbox[b * 4 + 2], nb3 = bbox[b * 4 + 3];
    const float x1 = fminf(fmaxf(nb0 * 64.f, 0.f), 63.f);
    const float y1 = fminf(fmaxf(nb1 * 64.f, 0.f), 63.f);
    const float x2 = fmaxf(fminf(nb2 * 64.f, 64.f), x1 + 1.f);
    const float y2 = fmaxf(fminf(nb3 * 64.f, 64.f), y1 + 1.f);
    const float roi_w = fmaxf(x2 - x1, 1.f);
    const float roi_h = fmaxf(y2 - y1, 1.f);
    const float gwf = ceilf(roi_w), ghf = ceilf(roi_h);
    const int   gw = (int)gwf,  gh = (int)ghf;           // adaptive grid dims
    const float bw = roi_w / gwf, bh = roi_h / ghf;      // sample spacing
    const float inv_n = 1.f / (gwf * ghf);
    const int   nsamp = gw * gh;                         // <= 4096

    int lymin = 64, lymax = -1, lxmin = 64, lxmax = -1;
    for (int s = lane; s < nsamp; s += 32) {
        const int iy = s / gw;
        const int ix = s - iy * gw;
        float y = fmaxf(y1 + ((float)iy + 0.5f) * bh, 0.f);   // y < 64
        float x = fmaxf(x1 + ((float)ix + 0.5f) * bw, 0.f);   // x < 64
        const int y0  = min((int)floorf(y), 63);
        const int x0  = min((int)floorf(x), 63);
        const int y1i = min(y0 + 1, 63);
        const int x1i = min(x0 + 1, 63);
        const float ly = y - (float)y0;
        const float lx = x - (float)x0;
        atomicAdd(&wmap[y0  * 64 + x0 ], (1.f - ly) * (1.f - lx));
        atomicAdd(&wmap[y0  * 64 + x1i], (1.f - ly) * lx);
        atomicAdd(&wmap[y1i * 64 + x0 ], ly * (1.f - lx));
        atomicAdd(&wmap[y1i * 64 + x1i], ly * lx);
        lymin = min(lymin, y0); lymax = max(lymax, y1i);
        lxmin = min(lxmin, x0); lxmax = max(lxmax, x1i);
    }
    if (lymax >= 0) {
        atomicMin(&sbnd[0], lymin); atomicMax(&sbnd[1], lymax);
        atomicMin(&sbnd[2], lxmin); atomicMax(&sbnd[3], lxmax);
    }
    __syncthreads();

    const int ry0 = sbnd[0], ry1 = sbnd[1], rx0 = sbnd[2], rx1 = sbnd[3];
    const int fw = rx1 - rx0 + 1;                 // footprint width
    const int P  = fw * (ry1 - ry0 + 1);          // footprint pixel count

    // ---------------- Phase B: WMMA weighted reduction ----------------------
    // out[c0+m] = sum_p feat[b, c0+m, p] * w[p]
    // A[16x4]: lane L<16 -> A[L, K=0..1]; lane L+16 -> A[L, K=2..3]
    // B[4x16]: broadcast w over columns; same half-wave K split as A.
    v8f acc = {};                                  // C/D accumulator
    const float* featb = feat + ((size_t)(b * CCH + c0)) * HWSZ;
    const int half = lane >> 4;                    // 0: lanes 0-15, 1: 16-31
    const int arow = lane & 15;                    // channel row within tile

    for (int pk = 0; pk < P; pk += CHUNK) {
        __syncthreads();
        // per-chunk pixel offsets + normalized weights
        for (int i = lane; i < CHUNK; i += 32) {
            const int p = pk + i;
            int off = 0; float w = 0.f;
            if (p < P) {
                const int yy = p / fw;
                const int xx = p - yy * fw;
                off = (ry0 + yy) * 64 + (rx0 + xx);
                w = wmap[off] * inv_n;
            }
            offs[i] = off;
            wchunk[i] = w;
        }
        __syncthreads();
        // stage 16 channels x CHUNK pixels, coalesced along pixels
        for (int e = lane; e < 16 * CHUNK; e += 32) {
            const int ch  = e >> 6;      // e / CHUNK
            const int pix = e & (CHUNK - 1);
            ftile[ch * ASTR + pix] = featb[(size_t)ch * HWSZ + offs[pix]];
        }
        __syncthreads();
        // 16 WMMA k-steps of K=4 over this chunk
        #pragma unroll 4
        for (int k0 = 0; k0 < CHUNK; k0 += 4) {
            const int kk = k0 + half * 2;
            v2f A = { ftile[arow * ASTR + kk], ftile[arow * ASTR + kk + 1] };
            v2f Bv = { wchunk[kk], wchunk[kk + 1] };
            acc = __builtin_amdgcn_wmma_f32_16x16x4_f32(
                false, A, false, Bv, (short)0, acc, false, false);
        }
    }

    // D: lane 0 holds M=0..7 (any column), lane 16 holds M=8..15
    if ((lane & 15) == 0) {
        float* ob = out + (size_t)b * CCH + c0 + half * 8;
        #pragma unroll
        for (int g = 0; g < 8; ++g) ob[g] = acc[g];
    }
}

extern "C" void kernel_launch(void* const* d_in, const int* in_sizes, int n_in,
                              void* d_out, int out_size, void* d_ws, size_t ws_size,
                              hipStream_t stream)
{
    const float* feat = (const float*)d_in[0];   // [256,256,64,64] f32
    const float* bbox = (const float*)d_in[1];   // [256,4] f32
    float* out = (float*)d_out;                  // [256,256] f32

    dim3 grid(CCH / 16, BBOX, 1);   // (channel tile, box)
    dim3 block(32, 1, 1);           // one wave32 per block
    roi_align_wmma_kernel<<<grid, block, 0, stream>>>(feat, bbox, out);
}